// AttentionManifold_43018392437194
// MI455X (gfx1250) — compile-verified
//
#include <hip/hip_runtime.h>
#include <math.h>

// MI455X / gfx1250, wave32. f32 WMMA (V_WMMA_F32_16X16X4_F32) for the GEMMs
// (precision-faithful to the f32 reference); GLOBAL_LOAD_ASYNC_TO_LDS_B128
// (ASYNCcnt) for the K-tile staging in the pair-energy kernel.

typedef float v2f __attribute__((ext_vector_type(2)));
typedef float v8f __attribute__((ext_vector_type(8)));

#define BS   4
#define MM_  64
#define DIN  128
#define DOUT 64
#define PP   16
#define LDA  17   // LDS row pitch for padded scratch
#define JW   8    // j-values (waves) per block in kernel 2

__device__ __forceinline__ float wave_sum32(float v) {
#pragma unroll
  for (int m = 16; m >= 1; m >>= 1) v += __shfl_xor(v, m, 32);
  return v;
}

// Async DMA of 16 bytes global -> LDS (one per lane). GV addressing, ASYNCcnt.
// Low 32 bits of a generic pointer to __shared__ == LDS byte offset
// (LDS aperture occupies addr[63:32]; addr[31:0] is the LDS offset).
__device__ __forceinline__ void async_copy_b128(void* lds_dst, const float* gsrc) {
  unsigned lds_off = (unsigned)(size_t)lds_dst;
  asm volatile("global_load_async_to_lds_b128 %0, %1, off"
               :: "v"(lds_off), "v"(gsrc) : "memory");
}
__device__ __forceinline__ void wait_async0() {
  asm volatile("s_wait_asynccnt 0" ::: "memory");
}

// ---------------------------------------------------------------------------
// Kernel 1: one wave per (b, m, t), t in {Q,K,V}.
//   A = W^T x[b,m] (64x16) via v_wmma_f32_16x16x4_f32, then in-wave MGS QR.
//   V only materialized at m = M-1 (whole-wave early exit keeps EXEC all-1s
//   for every executed WMMA).
// ---------------------------------------------------------------------------
__global__ __launch_bounds__(32) void k_frmap_qr(
    const float* __restrict__ x, const float* __restrict__ Wq,
    const float* __restrict__ Wk, const float* __restrict__ Wv,
    float* __restrict__ Qws, float* __restrict__ Kws, float* __restrict__ Vws)
{
  const int lane = threadIdx.x;
  const int wid  = blockIdx.x;            // 0 .. 3*BS*MM_-1
  const int t    = wid / (BS * MM_);      // 0=Q, 1=K, 2=V
  const int bm   = wid % (BS * MM_);
  const int b    = bm / MM_;
  const int m    = bm % MM_;
  if (t == 2 && m != MM_ - 1) return;     // uniform per wave

  const float* W  = (t == 0) ? Wq : (t == 1) ? Wk : Wv;
  const float* xb = x + (size_t)bm * DIN * PP;

  const int nl  = lane & 15;              // M (A) / N (B) index
  const int klo = (lane < 16) ? 0 : 2;    // K sub-lane split

  __shared__ float A[DOUT * LDA];

  v8f acc[4];
#pragma unroll
  for (int d = 0; d < 4; ++d) acc[d] = (v8f){0.f,0.f,0.f,0.f,0.f,0.f,0.f,0.f};

  // out[o][p] = sum_i W[i][o] x[i][p];  4 row-tiles of 16, K=128 in steps of 4
  for (int kk = 0; kk < DIN / 4; ++kk) {
    const int k0 = 4 * kk + klo;
    v2f bv;                               // B[k][n] = x[k][n]
    bv.x = xb[(k0    ) * PP + nl];
    bv.y = xb[(k0 + 1) * PP + nl];
#pragma unroll
    for (int d = 0; d < 4; ++d) {
      v2f av;                             // A[m][k] = W[k][d*16+m]
      av.x = W[(k0    ) * DOUT + d * 16 + nl];
      av.y = W[(k0 + 1) * DOUT + d * 16 + nl];
      acc[d] = __builtin_amdgcn_wmma_f32_16x16x4_f32(
          false, av, false, bv, (short)0, acc[d], false, false);
    }
  }

  // spill D-layout (VGPR r -> rows r / r+8) into LDS row-major [64][16]
#pragma unroll
  for (int d = 0; d < 4; ++d)
#pragma unroll
    for (int r = 0; r < 8; ++r) {
      const int row = d * 16 + r + ((lane < 16) ? 0 : 8);
      A[row * LDA + nl] = acc[d][r];
    }
  __syncthreads();

  // Modified Gram-Schmidt; column signs differ from Householder QR but all
  // downstream use is via projections QQ^T (sign-invariant).
  const int r0 = lane, r1 = lane + 32;    // each lane owns 2 rows
  for (int j = 0; j < PP; ++j) {
    float q0 = A[r0 * LDA + j], q1 = A[r1 * LDA + j];
    const float nrm = wave_sum32(q0 * q0 + q1 * q1);
    const float inv = 1.0f / sqrtf(fmaxf(nrm, 1e-30f));
    q0 *= inv; q1 *= inv;
    A[r0 * LDA + j] = q0; A[r1 * LDA + j] = q1;
    for (int l = j + 1; l < PP; ++l) {
      float u0 = A[r0 * LDA + l], u1 = A[r1 * LDA + l];
      const float dt = wave_sum32(q0 * u0 + q1 * u1);
      A[r0 * LDA + l] = u0 - dt * q0;
      A[r1 * LDA + l] = u1 - dt * q1;
    }
  }
  __syncthreads();

  float* dst = (t == 0) ? (Qws + (size_t)bm * DOUT * PP)
             : (t == 1) ? (Kws + (size_t)bm * DOUT * PP)
                        : (Vws + (size_t)b  * DOUT * PP);
#pragma unroll
  for (int p = 0; p < PP; ++p) {
    dst[r0 * PP + p] = A[r0 * LDA + p];
    dst[r1 * PP + p] = A[r1 * LDA + p];
  }
}

// ---------------------------------------------------------------------------
// Kernel 2: block = 8 waves = 8 consecutive j for one b. Each wave keeps its
// Q_j WMMA A-operands in registers (loaded once); the block async-DMAs each
// K_i tile (4 KB) into LDS once, shared by all 8 waves.
//   C = Q_j^T K_i (16x16) via WMMA (K=64); t2 = ||C||_F^2.
//   E = C C^T via WMMA (A and B operand registers are identical); t4=||E||_F^2.
//   tr(D^4) = 2(p - 2 t2 + t4);  score = 1/(1+log1p(sqrt(tr(D^4))))
// ---------------------------------------------------------------------------
__global__ __launch_bounds__(32 * JW) void k_energy(
    const float* __restrict__ Qws, const float* __restrict__ Kws,
    float* __restrict__ scores)
{
  const int b    = blockIdx.y;
  const int jblk = blockIdx.x;
  const int tid  = threadIdx.x;
  const int w    = tid >> 5;              // wave id -> j
  const int lane = tid & 31;
  const int j    = jblk * JW + w;
  const int nl   = lane & 15;
  const int klo  = (lane < 16) ? 0 : 2;

  __shared__ __align__(16) float Ksh[DOUT * PP];   // 4 KB K_i tile, row-major
  __shared__ float Csh[JW][16 * LDA];              // per-wave C scratch

  // Preload Q_j A-operands: qa[kk] covers K-rows 4kk+klo, 4kk+klo+1
  const float* Q = Qws + ((size_t)(b * MM_ + j)) * DOUT * PP;
  v2f qa[DOUT / 4];
#pragma unroll
  for (int kk = 0; kk < DOUT / 4; ++kk) {
    const int k0 = 4 * kk + klo;
    qa[kk].x = Q[(k0    ) * PP + nl];
    qa[kk].y = Q[(k0 + 1) * PP + nl];
  }

  for (int i = 0; i < MM_; ++i) {
    // Stage K_i: 256 threads x 16 B = 4 KB in one async shot (ASYNCcnt).
    const float* ksrc = Kws + ((size_t)(b * MM_ + i)) * DOUT * PP + tid * 4;
    async_copy_b128(&Ksh[tid * 4], ksrc);
    wait_async0();
    __syncthreads();

    v8f accC = (v8f){0.f,0.f,0.f,0.f,0.f,0.f,0.f,0.f};
#pragma unroll
    for (int kk = 0; kk < DOUT / 4; ++kk) {
      const int k0 = 4 * kk + klo;
      v2f bv;                             // B[k][n] = K[k][n]
      bv.x = Ksh[(k0    ) * PP + nl];
      bv.y = Ksh[(k0 + 1) * PP + nl];
      accC = __builtin_amdgcn_wmma_f32_16x16x4_f32(
          false, qa[kk], false, bv, (short)0, accC, false, false);
    }

    float loc = 0.f;
#pragma unroll
    for (int r = 0; r < 8; ++r) loc += accC[r] * accC[r];
    const float t2 = wave_sum32(loc);

    // C -> per-wave LDS scratch (in-wave LDS ordering; no block barrier needed)
#pragma unroll
    for (int r = 0; r < 8; ++r) {
      const int row = r + ((lane < 16) ? 0 : 8);
      Csh[w][row * LDA + nl] = accC[r];
    }

    v8f accE = (v8f){0.f,0.f,0.f,0.f,0.f,0.f,0.f,0.f};
#pragma unroll
    for (int kk = 0; kk < 4; ++kk) {
      const int k0 = 4 * kk + klo;
      v2f av;                             // same registers serve A and B for C*C^T
      av.x = Csh[w][nl * LDA + k0];
      av.y = Csh[w][nl * LDA + k0 + 1];
      accE = __builtin_amdgcn_wmma_f32_16x16x4_f32(
          false, av, false, av, (short)0, accE, false, false);
    }

    loc = 0.f;
#pragma unroll
    for (int r = 0; r < 8; ++r) loc += accE[r] * accE[r];
    const float t4 = wave_sum32(loc);

    if (lane == 0) {
      const float tr4    = 2.0f * ((float)PP - 2.0f * t2 + t4);
      const float energy = sqrtf(fmaxf(tr4, 0.0f));
      scores[((size_t)b * MM_ + j) * MM_ + i] = 1.0f / (1.0f + log1pf(energy));
    }
    __syncthreads();   // all waves done with Ksh before next i overwrites it
  }
}

// ---------------------------------------------------------------------------
// Kernel 3: one block per (b, j): softmax over i, take weight at i = M-1,
// scale Vp[b] = V V^T (V = QR(Wv^T x[b,M-1]).Q) into the output tile.
// ---------------------------------------------------------------------------
__global__ __launch_bounds__(256) void k_out(
    const float* __restrict__ scores, const float* __restrict__ Vws,
    float* __restrict__ out)
{
  const int j = blockIdx.x, b = blockIdx.y;
  const float* sc = scores + ((size_t)b * MM_ + j) * MM_;

  float mx = -1e30f;
  for (int i = 0; i < MM_; ++i) mx = fmaxf(mx, sc[i]);
  float s = 0.f;
  for (int i = 0; i < MM_; ++i) s += expf(sc[i] - mx);
  const float w = expf(sc[MM_ - 1] - mx) / s;

  const float* V = Vws + (size_t)b * DOUT * PP;
  float* o = out + ((size_t)b * MM_ + j) * DOUT * DOUT;
  for (int e = threadIdx.x; e < DOUT * DOUT; e += 256) {
    const int r = e >> 6, c = e & 63;
    float acc = 0.f;
#pragma unroll
    for (int p = 0; p < PP; ++p) acc += V[r * PP + p] * V[c * PP + p];
    o[e] = w * acc;
  }
}

// ---------------------------------------------------------------------------
extern "C" void kernel_launch(void* const* d_in, const int* in_sizes, int n_in,
                              void* d_out, int out_size, void* d_ws, size_t ws_size,
                              hipStream_t stream)
{
  (void)in_sizes; (void)n_in; (void)out_size; (void)ws_size;
  const float* x  = (const float*)d_in[0];
  const float* Wq = (const float*)d_in[1];
  const float* Wk = (const float*)d_in[2];
  const float* Wv = (const float*)d_in[3];

  float* ws     = (float*)d_ws;
  float* Qws    = ws;                                  // 4*64*64*16 floats
  float* Kws    = Qws + (size_t)BS * MM_ * DOUT * PP;  // 4*64*64*16
  float* Vws    = Kws + (size_t)BS * MM_ * DOUT * PP;  // 4*64*16 (m = M-1 only)
  float* scores = Vws + (size_t)BS * DOUT * PP;        // 4*64*64

  k_frmap_qr<<<dim3(3 * BS * MM_), dim3(32), 0, stream>>>(
      x, Wq, Wk, Wv, Qws, Kws, Vws);
  k_energy<<<dim3(MM_ / JW, BS), dim3(32 * JW), 0, stream>>>(Qws, Kws, scores);
  k_out<<<dim3(MM_, BS), dim3(256), 0, stream>>>(scores, Vws, (float*)d_out);
}